// Classifier_41162966565050
// MI455X (gfx1250) — compile-verified
//
#include <hip/hip_runtime.h>

// ---------------------------------------------------------------------------
// GraphSAGE (mean aggregator, 3 layers) + linear classifier + softmax
// Target: MI455X (gfx1250), wave32, fp32 WMMA 16x16x4.
// Layer GEMM: 5 row-tiles per block with register-resident weight fragments
// (weights loaded once per block instead of once per tile -> 5x less L2
// weight traffic).
// ---------------------------------------------------------------------------

typedef float v2f __attribute__((ext_vector_type(2)));
typedef float v8f __attribute__((ext_vector_type(8)));

#define N_NODES  50000
#define N_EDGES  600000
#define DIM      128
#define NCLS     47
#define NCLS_PAD 48
#define NLAYERS  3
#define LDS_STRIDE 132        // 128 + 4 pad: rows land 4 banks apart (64-bank LDS)
#define TILES_PER_BLOCK 5     // 3125 row tiles = 625 blocks * 5
#define KSTEPS (DIM / 4)      // 32 fp32-WMMA K-steps

// ---------------------------------------------------------------------------
// Degree / inverse-degree
// ---------------------------------------------------------------------------
__global__ __launch_bounds__(256)
void deg_kernel(const int* __restrict__ dst, float* __restrict__ deg) {
    int e = blockIdx.x * blockDim.x + threadIdx.x;
    if (e < N_EDGES) atomicAdd(&deg[dst[e]], 1.0f);
}

__global__ __launch_bounds__(256)
void invdeg_kernel(float* __restrict__ deg) {
    int i = blockIdx.x * blockDim.x + threadIdx.x;
    if (i < N_NODES) {
        float d = deg[i];
        deg[i] = 1.0f / fmaxf(d, 1.0f);
    }
}

// ---------------------------------------------------------------------------
// Scatter-add: one wave per edge. Each lane moves 4 floats (float4 load,
// 4x global_atomic_add_f32). h and h_neigh both live in the 192 MB L2.
// ---------------------------------------------------------------------------
__global__ __launch_bounds__(256)
void scatter_kernel(const float* __restrict__ h_in, const int* __restrict__ src,
                    const int* __restrict__ dst, float* __restrict__ h_neigh) {
    int wave = blockIdx.x * (blockDim.x >> 5) + (threadIdx.x >> 5);
    int lane = threadIdx.x & 31;
    if (wave >= N_EDGES) return;
    int s = src[wave];
    int d = dst[wave];
    const float4 v = *reinterpret_cast<const float4*>(h_in + (size_t)s * DIM + lane * 4);
    float* outp = h_neigh + (size_t)d * DIM + lane * 4;
    atomicAdd(outp + 0, v.x);
    atomicAdd(outp + 1, v.y);
    atomicAdd(outp + 2, v.z);
    atomicAdd(outp + 3, v.w);
}

// ---------------------------------------------------------------------------
// Fused SAGE layer: h_out = relu(h_in @ W_self + (h_neigh*inv_deg) @ W_neigh + b)
// Block = 256 threads = 8 waves; wave w owns N-columns [16w, 16w+16).
// Each block sweeps TILES_PER_BLOCK row-tiles of 16, reusing register-held
// B fragments (both weight matrices) across all tiles.
// A 16x4 fp32 frag layout: lane<16 -> K {k,k+1}; lane>=16 -> K {k+2,k+3}.
// C/D 16x16 fp32 layout: VGPR i -> (M = i + 8*(lane>=16), N = lane&15).
// ---------------------------------------------------------------------------
__global__ __launch_bounds__(256)
void layer_kernel(const float* __restrict__ h_in, const float* __restrict__ h_neigh,
                  const float* __restrict__ inv_deg,
                  const float* __restrict__ w_self, const float* __restrict__ w_neigh,
                  const float* __restrict__ bias, float* __restrict__ h_out) {
    __shared__ float lds_self[16 * LDS_STRIDE];
    __shared__ float lds_ngh[16 * LDS_STRIDE];

    const int tid  = threadIdx.x;
    const int wv   = tid >> 5;
    const int lane = tid & 31;
    const int colBase = wv * 16;
    const int mrow  = lane & 15;         // A-row / B-column index within tile
    const int khalf = (lane >> 4) * 2;   // lanes 0-15: K+{0,1}; lanes 16-31: K+{2,3}
    const int nCol  = colBase + mrow;

    // Register-resident B fragments for this wave's 16 output columns.
    v2f bS[KSTEPS], bN[KSTEPS];
    #pragma unroll
    for (int t = 0; t < KSTEPS; ++t) {
        int kk = t * 4 + khalf;
        bS[t].x = w_self [(size_t)kk * DIM + nCol];
        bS[t].y = w_self [(size_t)(kk + 1) * DIM + nCol];
        bN[t].x = w_neigh[(size_t)kk * DIM + nCol];
        bN[t].y = w_neigh[(size_t)(kk + 1) * DIM + nCol];
    }
    const float bb = bias[nCol];
    const int rOff = (lane >> 4) * 8;

    for (int tIdx = 0; tIdx < TILES_PER_BLOCK; ++tIdx) {
        const int rowBase = (blockIdx.x * TILES_PER_BLOCK + tIdx) * 16;

        // Stage A tiles: each thread loads 8 consecutive floats of each tile.
        {
            int idx = tid * 8;            // 256*8 = 2048 = 16*128 exactly
            int r = idx >> 7;
            int c = idx & 127;
            float sc = inv_deg[rowBase + r];
            const float4* ps = reinterpret_cast<const float4*>(h_in    + (size_t)(rowBase + r) * DIM + c);
            const float4* pn = reinterpret_cast<const float4*>(h_neigh + (size_t)(rowBase + r) * DIM + c);
            float4 s0 = ps[0], s1 = ps[1];
            float4 n0 = pn[0], n1 = pn[1];
            n0.x *= sc; n0.y *= sc; n0.z *= sc; n0.w *= sc;
            n1.x *= sc; n1.y *= sc; n1.z *= sc; n1.w *= sc;
            *reinterpret_cast<float4*>(&lds_self[r * LDS_STRIDE + c])     = s0;
            *reinterpret_cast<float4*>(&lds_self[r * LDS_STRIDE + c + 4]) = s1;
            *reinterpret_cast<float4*>(&lds_ngh [r * LDS_STRIDE + c])     = n0;
            *reinterpret_cast<float4*>(&lds_ngh [r * LDS_STRIDE + c + 4]) = n1;
        }
        __syncthreads();

        v8f acc = {};
        #pragma unroll
        for (int t = 0; t < KSTEPS; ++t) {
            int kk = t * 4 + khalf;
            v2f a_s, a_n;
            a_s.x = lds_self[mrow * LDS_STRIDE + kk];
            a_s.y = lds_self[mrow * LDS_STRIDE + kk + 1];
            a_n.x = lds_ngh [mrow * LDS_STRIDE + kk];
            a_n.y = lds_ngh [mrow * LDS_STRIDE + kk + 1];
            acc = __builtin_amdgcn_wmma_f32_16x16x4_f32(false, a_s, false, bS[t],
                                                        (short)0, acc, false, false);
            acc = __builtin_amdgcn_wmma_f32_16x16x4_f32(false, a_n, false, bN[t],
                                                        (short)0, acc, false, false);
        }

        #pragma unroll
        for (int i = 0; i < 8; ++i) {
            float v = acc[i] + bb;
            v = fmaxf(v, 0.0f);
            h_out[(size_t)(rowBase + rOff + i) * DIM + nCol] = v;
        }
        __syncthreads();   // A tiles consumed before next iteration overwrites
    }
}

// ---------------------------------------------------------------------------
// Classifier GEMM: logits[N,48] = h @ wc + bc  (cols >= 47 masked to 0 weight)
// One wave per 16x16 tile; grid = (3125 row tiles, 3 col tiles).
// ---------------------------------------------------------------------------
__global__ __launch_bounds__(32)
void cls_kernel(const float* __restrict__ h, const float* __restrict__ wc,
                const float* __restrict__ bc, float* __restrict__ logits) {
    const int rowBase = blockIdx.x * 16;
    const int colBase = blockIdx.y * 16;
    const int lane  = threadIdx.x & 31;
    const int mrow  = lane & 15;
    const int khalf = (lane >> 4) * 2;
    const int nCol  = colBase + mrow;
    const bool nOk  = (nCol < NCLS);

    v8f acc = {};
    #pragma unroll 4
    for (int k = 0; k < DIM; k += 4) {
        int kk = k + khalf;
        v2f a, b;
        a.x = h[(size_t)(rowBase + mrow) * DIM + kk];
        a.y = h[(size_t)(rowBase + mrow) * DIM + kk + 1];
        b.x = nOk ? wc[(size_t)kk * NCLS + nCol] : 0.0f;
        b.y = nOk ? wc[(size_t)(kk + 1) * NCLS + nCol] : 0.0f;
        acc = __builtin_amdgcn_wmma_f32_16x16x4_f32(false, a, false, b,
                                                    (short)0, acc, false, false);
    }

    float bb = nOk ? bc[nCol] : 0.0f;
    int rOff = (lane >> 4) * 8;
    #pragma unroll
    for (int i = 0; i < 8; ++i) {
        logits[(size_t)(rowBase + rOff + i) * NCLS_PAD + nCol] = acc[i] + bb;
    }
}

// ---------------------------------------------------------------------------
// Softmax over 47 classes: one wave per row (lane covers cols lane, lane+32).
// ---------------------------------------------------------------------------
__global__ __launch_bounds__(256)
void softmax_kernel(const float* __restrict__ logits, float* __restrict__ out) {
    const int wv   = threadIdx.x >> 5;
    const int lane = threadIdx.x & 31;
    const int row  = blockIdx.x * 8 + wv;

    const float* lrow = logits + (size_t)row * NCLS_PAD;
    float v0 = lrow[lane];                                        // lane <= 31 < 47
    float v1 = (lane + 32 < NCLS) ? lrow[lane + 32] : -__builtin_inff();

    float m = fmaxf(v0, v1);
    #pragma unroll
    for (int off = 16; off > 0; off >>= 1) m = fmaxf(m, __shfl_xor(m, off, 32));

    float e0 = __expf(v0 - m);
    float e1 = (lane + 32 < NCLS) ? __expf(v1 - m) : 0.0f;
    float s = e0 + e1;
    #pragma unroll
    for (int off = 16; off > 0; off >>= 1) s += __shfl_xor(s, off, 32);

    float inv = 1.0f / s;
    float* orow = out + (size_t)row * NCLS;
    orow[lane] = e0 * inv;
    if (lane + 32 < NCLS) orow[lane + 32] = e1 * inv;
}

// ---------------------------------------------------------------------------
// Host launcher
// ---------------------------------------------------------------------------
extern "C" void kernel_launch(void* const* d_in, const int* in_sizes, int n_in,
                              void* d_out, int out_size, void* d_ws, size_t ws_size,
                              hipStream_t stream) {
    const float* features = (const float*)d_in[0];
    const int*   src      = (const int*)  d_in[1];
    const int*   dst      = (const int*)  d_in[2];
    const float* w_self   = (const float*)d_in[3];
    const float* w_neigh  = (const float*)d_in[4];
    const float* b        = (const float*)d_in[5];
    const float* wc       = (const float*)d_in[6];
    const float* bc       = (const float*)d_in[7];
    float* out = (float*)d_out;

    // Workspace layout (floats): inv_deg | hA | hB | h_neigh | logits
    float* ws      = (float*)d_ws;
    float* inv_deg = ws;                                    // N (deg, then 1/deg in place)
    float* hA      = ws + 50048;                            // N*D
    float* hB      = hA + (size_t)N_NODES * DIM;            // N*D
    float* h_ngh   = hB + (size_t)N_NODES * DIM;            // N*D
    float* logits  = h_ngh + (size_t)N_NODES * DIM;         // N*48

    // Degrees -> inverse degrees (clamped at 1).
    hipMemsetAsync(inv_deg, 0, N_NODES * sizeof(float), stream);
    deg_kernel<<<(N_EDGES + 255) / 256, 256, 0, stream>>>(dst, inv_deg);
    invdeg_kernel<<<(N_NODES + 255) / 256, 256, 0, stream>>>(inv_deg);

    const float* h_cur = features;
    float* bufs[2] = { hA, hB };
    const int layerBlocks = N_NODES / (16 * TILES_PER_BLOCK);   // 625
    for (int l = 0; l < NLAYERS; ++l) {
        hipMemsetAsync(h_ngh, 0, (size_t)N_NODES * DIM * sizeof(float), stream);
        scatter_kernel<<<N_EDGES / 8, 256, 0, stream>>>(h_cur, src, dst, h_ngh);
        float* h_next = bufs[l & 1];
        layer_kernel<<<layerBlocks, 256, 0, stream>>>(
            h_cur, h_ngh, inv_deg,
            w_self + (size_t)l * DIM * DIM,
            w_neigh + (size_t)l * DIM * DIM,
            b + (size_t)l * DIM,
            h_next);
        h_cur = h_next;
    }

    dim3 cgrid(N_NODES / 16, NCLS_PAD / 16);
    cls_kernel<<<cgrid, 32, 0, stream>>>(h_cur, wc, bc, logits);
    softmax_kernel<<<N_NODES / 8, 256, 0, stream>>>(logits, out);
}